// Second_order_conv2d_24000277250380
// MI455X (gfx1250) — compile-verified
//
#include <hip/hip_runtime.h>

typedef float v2f __attribute__((ext_vector_type(2)));
typedef float v8f __attribute__((ext_vector_type(8)));

#define BATCH 8
#define C_IN  64
#define HW    96
#define OC    128
#define NF    54     // features per channel
#define NFP   56     // padded to multiple of 4 for K=4 WMMA stepping
#define NPAIR 28     // NFP/2 feature pairs (pair 27 = zero padding f=54,55)
#define NSUB  6      // 6 x 16 = 96 pixels (one image row) per workgroup
#define NPAD  112    // pixel-dim stride: NPAD*2 % 64 == 32 -> half-waves on disjoint bank halves

// PM_cross(2,9) pair ordering: for j=1..8, pairs (i, i+j), i=0..8-j
__device__ constexpr int PA[36] = {0,1,2,3,4,5,6,7, 0,1,2,3,4,5,6, 0,1,2,3,4,5,
                                   0,1,2,3,4, 0,1,2,3, 0,1,2, 0,1, 0};
__device__ constexpr int PB[36] = {1,2,3,4,5,6,7,8, 2,3,4,5,6,7,8, 3,4,5,6,7,8,
                                   4,5,6,7,8, 5,6,7,8, 6,7,8, 7,8, 8};

// Build the 54 x 96 second-order feature tile for one channel plane directly
// from global memory (clamped addresses + mask -> branch-free), pair-interleaved
// into LDS. Threads 0..191 participate: group 0 writes f=0..26, group 1 f=27..53.
static __device__ __forceinline__ void build_feat(float (*featbuf)[NPAD][2],
                                                  const float* __restrict__ xplane,
                                                  int y, int tid) {
  if (tid < 2 * HW) {
    const int grp = tid / HW;
    const int n   = tid - grp * HW;       // pixel within row
    float p[9];
#pragma unroll
    for (int kh = 0; kh < 3; ++kh) {
      const int  yy  = y + kh - 1;
      const bool oky = (yy >= 0) & (yy < HW);
      const int  yyc = yy < 0 ? 0 : (yy >= HW ? HW - 1 : yy);
#pragma unroll
      for (int kw = 0; kw < 3; ++kw) {
        const int  xx  = n + kw - 1;
        const bool okx = (xx >= 0) & (xx < HW);
        const int  xxc = xx < 0 ? 0 : (xx >= HW ? HW - 1 : xx);
        const float v  = xplane[yyc * HW + xxc];   // always in-bounds
        p[kh * 3 + kw] = (oky & okx) ? v : 0.0f;   // v_cndmask, no branch
      }
    }
    if (grp == 0) {
#pragma unroll
      for (int k = 0; k < 9; ++k) featbuf[k >> 1][n][k & 1] = p[k];
#pragma unroll
      for (int k = 0; k < 9; ++k) {
        const int f = 9 + k;
        featbuf[f >> 1][n][f & 1] = p[k] * p[k];
      }
#pragma unroll
      for (int g = 0; g < 9; ++g) {
        const int f = 18 + g;
        featbuf[f >> 1][n][f & 1] = p[PA[g]] * p[PB[g]];
      }
    } else {
#pragma unroll
      for (int g = 9; g < 36; ++g) {
        const int f = 18 + g;
        featbuf[f >> 1][n][f & 1] = p[PA[g]] * p[PB[g]];
      }
    }
  }
}

__global__ __launch_bounds__(256)
void Second_order_conv2d_kernel(const float* __restrict__ x,
                                const float* __restrict__ w,
                                float* __restrict__ out) {
  // double-buffered pair-interleaved features: [buf][f/2][pixel][f&1]
  __shared__ float featP[2][NPAIR][NPAD][2];

  const int y    = blockIdx.x;              // image row
  const int b    = blockIdx.y;              // batch
  const int tid  = threadIdx.x;
  const int wave = tid >> 5;
  const int lane = tid & 31;
  const int m     = lane & 15;              // output-channel row / pixel within subtile
  const int khalf = lane >> 4;              // K pair select: {0,1} vs {2,3}
  const int obase = wave * 16;              // this wave's 16 output channels

  // zero the padding feature pair (f = 54,55) in both buffers; never rewritten
  for (int i = tid; i < 2 * NPAD * 2; i += 256) {
    const int bufi = i / (NPAD * 2);
    const int j    = i - bufi * (NPAD * 2);
    featP[bufi][NPAIR - 1][j >> 1][j & 1] = 0.0f;
  }

  v8f acc[NSUB];
  const v8f vzero = {0.f, 0.f, 0.f, 0.f, 0.f, 0.f, 0.f, 0.f};
#pragma unroll
  for (int t = 0; t < NSUB; ++t) acc[t] = vzero;

  // weight row for this lane's output channel: w[o][c][f], o = obase + m
  const float* wbase  = w + (size_t)(obase + m) * (C_IN * NF);
  const float* xbatch = x + (size_t)b * C_IN * (HW * HW);

  // prologue: build channel 0's features
  build_feat(featP[0], xbatch, y, tid);
  __syncthreads();

  for (int c = 0; c < C_IN; ++c) {
    const int buf = c & 1;

    // ---- overlap: build channel c+1 into the other buffer (no barrier between) ----
    if (c + 1 < C_IN)
      build_feat(featP[buf ^ 1], xbatch + (size_t)(c + 1) * (HW * HW), y, tid);

    // prefetch channel c+2's halo rows (3 rows x 3 cachelines)
    if (c + 2 < C_IN && tid < 9) {
      const int r  = tid / 3;
      const int sg = tid - r * 3;
      int yy = y + r - 1;
      yy = yy < 0 ? 0 : (yy >= HW ? HW - 1 : yy);
      __builtin_prefetch(xbatch + (size_t)(c + 2) * (HW * HW) + yy * HW + sg * 32, 0, 0);
    }

    // ---- fp32 WMMA K-sweep on buffer `buf`: 14 K-steps x 6 subtiles, A reused 6x ----
    const float* wc = wbase + c * NF;
#pragma unroll
    for (int fb = 0; fb < NFP; fb += 4) {
      const int f0 = fb + 2 * khalf;        // this half-wave's K pair base
      v2f av;
      if (f0 < NF) {
        av = *reinterpret_cast<const v2f*>(wc + f0);  // even offset -> 8B aligned
      } else {
        av.x = 0.0f; av.y = 0.0f;           // K padding (f=54,55)
      }
      const int fp = (fb >> 1) + khalf;     // feature pair row
#pragma unroll
      for (int t = 0; t < NSUB; ++t) {
        // single aligned b64: {feat[f0][n], feat[f0+1][n]} -> contiguous VGPR pair
        v2f bv = *reinterpret_cast<const v2f*>(&featP[buf][fp][t * 16 + m][0]);
        acc[t] = __builtin_amdgcn_wmma_f32_16x16x4_f32(
            false, av, false, bv, (short)0, acc[t], false, false);
      }
    }

    __syncthreads();  // one barrier per channel: build(c+1) and sweep(c) both done
  }

  // ---- store D: VGPR v -> M = v (lanes 0-15) / v+8 (lanes 16-31), N = lane&15 ----
  float* outb = out + ((size_t)b * OC * HW + y) * HW;  // out[b][.][y][.]
#pragma unroll
  for (int t = 0; t < NSUB; ++t) {
#pragma unroll
    for (int v = 0; v < 8; ++v) {
      int o = obase + v + khalf * 8;
      outb[(size_t)o * (HW * HW) + t * 16 + m] = acc[t][v];
    }
  }
}

extern "C" void kernel_launch(void* const* d_in, const int* in_sizes, int n_in,
                              void* d_out, int out_size, void* d_ws, size_t ws_size,
                              hipStream_t stream) {
  const float* x = (const float*)d_in[0];   // [8, 64, 96, 96] fp32
  const float* w = (const float*)d_in[1];   // [128, 64, 54, 1] fp32
  float* out = (float*)d_out;               // [8, 128, 96, 96] fp32
  (void)in_sizes; (void)n_in; (void)out_size; (void)d_ws; (void)ws_size;

  dim3 grid(HW, BATCH);   // one block per (image row, batch)
  Second_order_conv2d_kernel<<<grid, 256, 0, stream>>>(x, w, out);
}